// PoseEstimator_89739046683477
// MI455X (gfx1250) — compile-verified
//
#include <hip/hip_runtime.h>
#include <math.h>

#define N_PTS 1024
#define M_PTS 1024
#define N_HYP 128
#define ERR_TH2 64.0f         // (8 px)^2
#define ERR_TH2_BITS 0x42800000   // bit pattern of 64.0f
#define INLIER_TH 0.25f
#define EPS 1e-8f

typedef __attribute__((ext_vector_type(2))) float v2f;
typedef __attribute__((ext_vector_type(8))) float v8f;

// ---- workspace layout (bytes) ----
#define WS_ROWSUM   0u        // 1024 f32
#define WS_SI0      4096u     // 512 i32
#define WS_SI1      6144u     // 512 i32
#define WS_AUGB     8192u     // 1024*4 f32
#define WS_INLCNT   24576u    // 128 i32
#define WS_VALCNT   25088u    // 1 i32
#define WS_TOTINL   25092u    // 1 i32
#define WS_PCNT     25096u    // 1 i32
#define WS_PLIST    25600u    // 128 i32
#define WS_LOSSPART 26624u    // 512 f32
#define WS_AUGA     32768u    // 128*1024*4 f32 = 2 MB

// ---------------- init accumulators (ws is poisoned, must self-init) -------
__global__ void k_init(int* inlCnt, int* valCnt, int* totInl) {
    int t = threadIdx.x;
    if (t < N_HYP) inlCnt[t] = 0;
    if (t == 0) { *valCnt = 0; *totInl = 0; }
}

// ---------------- augB (B-matrix columns) + valid count --------------------
__global__ void k_prep(const float* __restrict__ pts1, const int* __restrict__ indices0,
                       float* __restrict__ augB, int* valCnt) {
    int i = blockIdx.x * 256 + threadIdx.x;       // 0..1023
    float qx = pts1[2 * i], qy = pts1[2 * i + 1];
    augB[4 * i + 0] = -2.0f * qx;
    augB[4 * i + 1] = -2.0f * qy;
    augB[4 * i + 2] = 1.0f;
    augB[4 * i + 3] = qx * qx + qy * qy;
    __shared__ int sh[256];
    sh[threadIdx.x] = (indices0[i] > -1) ? 1 : 0;
    __syncthreads();
    for (int off = 128; off > 0; off >>= 1) {
        if (threadIdx.x < off) sh[threadIdx.x] += sh[threadIdx.x + off];
        __syncthreads();
    }
    if (threadIdx.x == 0) atomicAdd(valCnt, sh[0]);
}

// ---------------- per-row probability sums for categorical sampling -------
__global__ void k_rowsum(const float* __restrict__ scores, float* __restrict__ rowSum) {
    int r = blockIdx.x, t = threadIdx.x;
    float s = 0.0f;
    for (int k = t; k < M_PTS; k += 256) s += fabsf(scores[(size_t)r * M_PTS + k]) + EPS;
    __shared__ float sh[256];
    sh[t] = s; __syncthreads();
    for (int off = 128; off > 0; off >>= 1) {
        if (t < off) sh[t] += sh[t + off];
        __syncthreads();
    }
    if (t == 0) rowSum[r] = sh[0];
}

// ---------------- inverse-CDF categorical sampling (512 samples) ----------
__global__ void k_sample(const float* __restrict__ scores, const float* __restrict__ rowSum,
                         int* __restrict__ si0, int* __restrict__ si1) {
    __shared__ float cdf[N_PTS];
    int t = threadIdx.x;
    if (t == 0) {
        float run = 0.0f;
        for (int r = 0; r < N_PTS; r++) { run += rowSum[r]; cdf[r] = run; }
    }
    __syncthreads();
    float total = cdf[N_PTS - 1];
    // splitmix-style hash RNG, seeded deterministically (key 42)
    unsigned h = (unsigned)t * 0x9E3779B9u + 42u * 0x7F4A7C15u + 0x6C62272Eu;
    h ^= h >> 16; h *= 0x85EBCA6Bu; h ^= h >> 13; h *= 0xC2B2AE35u; h ^= h >> 16;
    float u = (float)(h >> 8) * (1.0f / 16777216.0f);
    if (u > 0.999999f) u = 0.999999f;
    float target = u * total;
    int lo = 0, hi = N_PTS - 1;
    while (lo < hi) { int mid = (lo + hi) >> 1; if (cdf[mid] > target) hi = mid; else lo = mid + 1; }
    int r = lo;
    float resid = target - (r > 0 ? cdf[r - 1] : 0.0f);
    float cum = 0.0f; int col = M_PTS - 1;
    for (int m = 0; m < M_PTS; m++) {
        cum += fabsf(scores[(size_t)r * M_PTS + m]) + EPS;
        if (cum > resid) { col = m; break; }
    }
    si0[t] = r; si1[t] = col;
}

// ---------------- 4-point normalized DLT, null-space via 9x9 Jacobi -------
__global__ void k_dlt(const float* __restrict__ pts0, const float* __restrict__ pts1,
                      const int* __restrict__ si0, const int* __restrict__ si1,
                      float* __restrict__ outH) {
    int hyp = blockIdx.x * blockDim.x + threadIdx.x;
    if (hyp >= N_HYP) return;
    float p0x[4], p0y[4], p1x[4], p1y[4];
    for (int j = 0; j < 4; j++) {
        int a = si0[hyp * 4 + j], b = si1[hyp * 4 + j];
        p0x[j] = pts0[2 * a]; p0y[j] = pts0[2 * a + 1];
        p1x[j] = pts1[2 * b]; p1y[j] = pts1[2 * b + 1];
    }
    float mx0 = 0, my0 = 0, mx1 = 0, my1 = 0;
    for (int j = 0; j < 4; j++) { mx0 += p0x[j]; my0 += p0y[j]; mx1 += p1x[j]; my1 += p1y[j]; }
    mx0 *= 0.25f; my0 *= 0.25f; mx1 *= 0.25f; my1 *= 0.25f;
    float d0 = 0, d1 = 0;
    for (int j = 0; j < 4; j++) {
        float ax = p0x[j] - mx0, ay = p0y[j] - my0;
        float bx = p1x[j] - mx1, by = p1y[j] - my1;
        d0 += sqrtf(ax * ax + ay * ay);
        d1 += sqrtf(bx * bx + by * by);
    }
    d0 *= 0.25f; d1 *= 0.25f;
    float s0 = 1.41421356f / fmaxf(d0, 1e-8f);
    float s1 = 1.41421356f / fmaxf(d1, 1e-8f);
    float x[4], y[4], xp[4], yp[4];
    for (int j = 0; j < 4; j++) {
        x[j]  = (p0x[j] - mx0) * s0; y[j]  = (p0y[j] - my0) * s0;
        xp[j] = (p1x[j] - mx1) * s1; yp[j] = (p1y[j] - my1) * s1;
    }
    // normal matrix Mm = A^T A  (A is 8x9)
    float Mm[9][9];
    for (int r = 0; r < 9; r++) for (int c = 0; c < 9; c++) Mm[r][c] = 0.0f;
    for (int j = 0; j < 4; j++) {
        float ax[9] = {0.f, 0.f, 0.f, -x[j], -y[j], -1.f, yp[j] * x[j], yp[j] * y[j], yp[j]};
        float ay[9] = {x[j], y[j], 1.f, 0.f, 0.f, 0.f, -xp[j] * x[j], -xp[j] * y[j], -xp[j]};
        for (int r = 0; r < 9; r++)
            for (int c = 0; c < 9; c++)
                Mm[r][c] += ax[r] * ax[c] + ay[r] * ay[c];
    }
    float V[9][9];
    for (int r = 0; r < 9; r++) for (int c = 0; c < 9; c++) V[r][c] = (r == c) ? 1.0f : 0.0f;
    // cyclic Jacobi sweeps for symmetric eigen-decomposition
    for (int sweep = 0; sweep < 10; sweep++) {
        for (int pp = 0; pp < 8; pp++) {
            for (int q = pp + 1; q < 9; q++) {
                float apq = Mm[pp][q];
                if (fabsf(apq) < 1e-20f) continue;
                float theta = 0.5f * (Mm[q][q] - Mm[pp][pp]) / apq;
                float tt = ((theta >= 0.f) ? 1.f : -1.f) / (fabsf(theta) + sqrtf(theta * theta + 1.f));
                float c = 1.0f / sqrtf(tt * tt + 1.0f);
                float sn = tt * c;
                for (int k = 0; k < 9; k++) {
                    float a1 = Mm[pp][k], a2 = Mm[q][k];
                    Mm[pp][k] = c * a1 - sn * a2; Mm[q][k] = sn * a1 + c * a2;
                }
                for (int k = 0; k < 9; k++) {
                    float a1 = Mm[k][pp], a2 = Mm[k][q];
                    Mm[k][pp] = c * a1 - sn * a2; Mm[k][q] = sn * a1 + c * a2;
                }
                for (int k = 0; k < 9; k++) {
                    float a1 = V[k][pp], a2 = V[k][q];
                    V[k][pp] = c * a1 - sn * a2; V[k][q] = sn * a1 + c * a2;
                }
            }
        }
    }
    int best = 0;
    for (int i = 1; i < 9; i++) if (Mm[i][i] < Mm[best][best]) best = i;
    float hv[9];
    for (int k = 0; k < 9; k++) hv[k] = V[k][best];
    // H = inv(T1) * Hn * T0
    float t0x = -s0 * mx0, t0y = -s0 * my0;
    float G[3][3];
    for (int r = 0; r < 3; r++) {
        float h0 = hv[3 * r + 0], h1 = hv[3 * r + 1], h2 = hv[3 * r + 2];
        G[r][0] = h0 * s0; G[r][1] = h1 * s0; G[r][2] = h0 * t0x + h1 * t0y + h2;
    }
    float is1 = 1.0f / s1;
    float Hm[3][3];
    for (int c = 0; c < 3; c++) {
        Hm[0][c] = is1 * G[0][c] + mx1 * G[2][c];
        Hm[1][c] = is1 * G[1][c] + my1 * G[2][c];
        Hm[2][c] = G[2][c];
    }
    float denom = Hm[2][2] + 1e-12f;
    for (int r = 0; r < 3; r++)
        for (int c = 0; c < 3; c++)
            outH[hyp * 9 + r * 3 + c] = Hm[r][c] / denom;
}

// ---------------- project pts0 under all H, build augA, count inliers -----
__global__ void k_project(const float* __restrict__ pts0, const float* __restrict__ pts1,
                          const int* __restrict__ indices0, const float* __restrict__ outH,
                          float* __restrict__ augA, int* __restrict__ inlCnt) {
    int n = blockIdx.x * 256 + threadIdx.x;
    int p = blockIdx.y;
    const float* H = outH + p * 9;
    float xx = pts0[2 * n], yy = pts0[2 * n + 1];
    float w = H[6] * xx + H[7] * yy + H[8];
    float iw = 1.0f / (w + 1e-12f);
    float px = (H[0] * xx + H[1] * yy + H[2]) * iw;
    float py = (H[3] * xx + H[4] * yy + H[5]) * iw;
    float* a = augA + ((size_t)p * N_PTS + n) * 4;
    a[0] = px; a[1] = py; a[2] = px * px + py * py; a[3] = 1.0f;
    int mi = indices0[n];
    int mc = (mi > 0) ? mi : 0;
    float qx = pts1[2 * mc], qy = pts1[2 * mc + 1];
    float dx = px - qx, dy = py - qy;
    int inl = ((mi > -1) && (dx * dx + dy * dy <= ERR_TH2)) ? 1 : 0;
    __shared__ int sh[256];
    sh[threadIdx.x] = inl; __syncthreads();
    for (int off = 128; off > 0; off >>= 1) {
        if (threadIdx.x < off) sh[threadIdx.x] += sh[threadIdx.x + off];
        __syncthreads();
    }
    if (threadIdx.x == 0) atomicAdd(&inlCnt[p], sh[0]);
}

// ------- ratio + total inliers + compacted list of surviving hypotheses ---
__global__ void k_ratio(const int* __restrict__ inlCnt, const int* __restrict__ valCnt,
                        int* __restrict__ totInl, float* __restrict__ outRatio,
                        int* __restrict__ pList, int* __restrict__ pCnt) {
    int t = threadIdx.x;  // 128 threads
    int vc = *valCnt;
    float nG = (float)(vc > 1 ? vc : 1);
    int c = inlCnt[t];
    float r = (float)c / nG;
    int keep = (r > INLIER_TH) ? 1 : 0;
    outRatio[t] = keep ? r : 0.0f;
    // inclusive prefix scan of keep flags (Hillis-Steele in LDS)
    __shared__ int sc[128];
    sc[t] = keep; __syncthreads();
    for (int off = 1; off < 128; off <<= 1) {
        int v = sc[t];
        int add = (t >= off) ? sc[t - off] : 0;
        __syncthreads();
        sc[t] = v + add;
        __syncthreads();
    }
    if (keep) pList[sc[t] - 1] = t;
    if (t == 127) *pCnt = sc[127];
    // total inlier count (denominator of the loss)
    __shared__ int si[128];
    si[t] = c; __syncthreads();
    for (int off = 64; off > 0; off >>= 1) {
        if (t < off) si[t] += si[t + off];
        __syncthreads();
    }
    if (t == 0) *totInl = si[0];
}

// ---------------- supp via f32 WMMA (augmented K=4 distance GEMM) ---------
// d2[p,n,m] = [px,py,|p|^2,1] . [-2qx,-2qy,1,|q|^2] ; one 16x16 tile / WMMA.
// supp[n,m] = (min over surviving p of d2) <= ERR_TH2.
// 8 m-tiles per wave: one A-fragment load feeds 8 WMMAs (register reuse),
// register double-buffer hides L2 latency. The running min is kept as an
// integer min on f32 bit patterns (monotone for d2 >= 0, sign-bit cases fall
// through the threshold test correctly) -> single v_min_i32 per element,
// no fminf NaN-canonicalization traffic.
__global__ void __launch_bounds__(32) k_supp(const float* __restrict__ augA,
                                             const float* __restrict__ augB,
                                             const int* __restrict__ pList,
                                             const int* __restrict__ pCnt,
                                             float* __restrict__ supp) {
    int mt8   = blockIdx.x;   // 8 groups of 8 m-tiles (128 columns)
    int ntile = blockIdx.y;   // 64 n-tiles
    int lane  = threadIdx.x;
    int kh  = lane >> 4;      // K half: 0 -> K=0,1 ; 1 -> K=2,3
    int l16 = lane & 15;      // row (A) / column (B) within a 16-wide tile
    v2f bf[8];
#pragma unroll
    for (int tb = 0; tb < 8; tb++)
        bf[tb] = *(const v2f*)(augB + ((size_t)(mt8 * 128 + tb * 16 + l16) * 4 + kh * 2));
    int dminI[8][8];
#pragma unroll
    for (int tb = 0; tb < 8; tb++)
#pragma unroll
        for (int i = 0; i < 8; i++) dminI[tb][i] = 0x7F7FFFFF;   // +FLT_MAX bits

    const float* abase = augA + (size_t)(ntile * 16 + l16) * 4 + kh * 2;
    int cnt = *pCnt;
    if (cnt > 0) {
        v2f aCur = *(const v2f*)(abase + (size_t)pList[0] * (N_PTS * 4));
        for (int j = 0; j < cnt; j++) {
            v2f aNext = aCur;
            if (j + 1 < cnt) {
                int pn = pList[j + 1];
                aNext = *(const v2f*)(abase + (size_t)pn * (N_PTS * 4));
            }
            v8f cz = {0, 0, 0, 0, 0, 0, 0, 0};
#pragma unroll
            for (int tb = 0; tb < 8; tb++) {
                v8f d2 = __builtin_amdgcn_wmma_f32_16x16x4_f32(
                    /*neg_a=*/false, aCur, /*neg_b=*/false, bf[tb],
                    /*c_mod=*/(short)0, cz, /*reuse_a=*/false, /*reuse_b=*/false);
#pragma unroll
                for (int i = 0; i < 8; i++) {
                    int b = __float_as_int(d2[i]);
                    dminI[tb][i] = (b < dminI[tb][i]) ? b : dminI[tb][i];
                }
            }
            aCur = aNext;
        }
    }
#pragma unroll
    for (int tb = 0; tb < 8; tb++) {
        int m = mt8 * 128 + tb * 16 + l16;
#pragma unroll
        for (int i = 0; i < 8; i++) {
            int n = ntile * 16 + kh * 8 + i;  // C/D layout: VGPR i -> rows i / i+8
            supp[(size_t)n * M_PTS + m] = (dminI[tb][i] <= ERR_TH2_BITS) ? 1.0f : 0.0f;
        }
    }
}

// ---------------- loss partials (fixed-order two-stage reduction) ---------
__global__ void k_losspart(const float* __restrict__ augA, const float* __restrict__ pts1,
                           const int* __restrict__ indices0, const float* __restrict__ scores,
                           const float* __restrict__ ratio, float* __restrict__ lossPart) {
    int n = blockIdx.x * 256 + threadIdx.x;
    int p = blockIdx.y;
    int mi = indices0[n];
    int mc = (mi > 0) ? mi : 0;
    const float* a = augA + ((size_t)p * N_PTS + n) * 4;
    float px = a[0], py = a[1];
    float qx = pts1[2 * mc], qy = pts1[2 * mc + 1];
    float dx = px - qx, dy = py - qy;
    bool mask = (mi > -1) && (dx * dx + dy * dy <= ERR_TH2);
    float lp = logf(fabsf(scores[(size_t)n * M_PTS + mc]) + EPS);
    float contrib = mask ? (-lp * ratio[p]) : 0.0f;
    __shared__ float sh[256];
    sh[threadIdx.x] = contrib; __syncthreads();
    for (int off = 128; off > 0; off >>= 1) {
        if (threadIdx.x < off) sh[threadIdx.x] += sh[threadIdx.x + off];
        __syncthreads();
    }
    if (threadIdx.x == 0) lossPart[p * gridDim.x + blockIdx.x] = sh[0];
}

__global__ void k_lossfinal(const float* __restrict__ lossPart, const int* __restrict__ totInl,
                            float* __restrict__ outLoss) {
    __shared__ float sh[512];
    int t = threadIdx.x;
    sh[t] = lossPart[t]; __syncthreads();
    for (int off = 256; off > 0; off >>= 1) {
        if (t < off) sh[t] += sh[t + off];
        __syncthreads();
    }
    if (t == 0) {
        int ti = *totInl;
        float denomv = (float)(ti > 1 ? ti : 1);
        *outLoss = sh[0] / denomv;
    }
}

extern "C" void kernel_launch(void* const* d_in, const int* in_sizes, int n_in,
                              void* d_out, int out_size, void* d_ws, size_t ws_size,
                              hipStream_t stream) {
    (void)in_sizes; (void)n_in; (void)out_size; (void)ws_size;
    const float* pts0     = (const float*)d_in[0];
    const float* pts1     = (const float*)d_in[1];
    const float* scores   = (const float*)d_in[2];
    const int*   indices0 = (const int*)d_in[3];

    float* outH     = (float*)d_out;                 // [128,3,3]
    float* outRatio = outH + N_HYP * 9;              // [128]
    float* outSupp  = outRatio + N_HYP;              // [1024,1024]
    float* outLoss  = outSupp + (size_t)N_PTS * M_PTS;  // scalar

    char* ws = (char*)d_ws;
    float* rowSum   = (float*)(ws + WS_ROWSUM);
    int*   si0      = (int*)(ws + WS_SI0);
    int*   si1      = (int*)(ws + WS_SI1);
    float* augB     = (float*)(ws + WS_AUGB);
    int*   inlCnt   = (int*)(ws + WS_INLCNT);
    int*   valCnt   = (int*)(ws + WS_VALCNT);
    int*   totInl   = (int*)(ws + WS_TOTINL);
    int*   pCnt     = (int*)(ws + WS_PCNT);
    int*   pList    = (int*)(ws + WS_PLIST);
    float* lossPart = (float*)(ws + WS_LOSSPART);
    float* augA     = (float*)(ws + WS_AUGA);

    k_init<<<1, 256, 0, stream>>>(inlCnt, valCnt, totInl);
    k_prep<<<4, 256, 0, stream>>>(pts1, indices0, augB, valCnt);
    k_rowsum<<<N_PTS, 256, 0, stream>>>(scores, rowSum);
    k_sample<<<1, 512, 0, stream>>>(scores, rowSum, si0, si1);
    k_dlt<<<1, 128, 0, stream>>>(pts0, pts1, si0, si1, outH);
    dim3 gp(4, N_HYP);
    k_project<<<gp, 256, 0, stream>>>(pts0, pts1, indices0, outH, augA, inlCnt);
    k_ratio<<<1, 128, 0, stream>>>(inlCnt, valCnt, totInl, outRatio, pList, pCnt);
    dim3 gs(M_PTS / 128, N_PTS / 16);
    k_supp<<<gs, 32, 0, stream>>>(augA, augB, pList, pCnt, outSupp);
    dim3 gl(4, N_HYP);
    k_losspart<<<gl, 256, 0, stream>>>(augA, pts1, indices0, scores, outRatio, lossPart);
    k_lossfinal<<<1, 512, 0, stream>>>(lossPart, totInl, outLoss);
}